// FLC_Pooling_conv_blurred_23347442221125
// MI455X (gfx1250) — compile-verified
//
#include <hip/hip_runtime.h>
#include <math.h>

// ---------------------------------------------------------------------------
// FLC pooling via DFT-as-GEMM on CDNA5 WMMA (f32 16x16x4), LDS-staged tiles.
//   G   = A  · x · Aᵀ          A: 160x128 complex (reflect-pad folded in, /321²)
//   low = |B64 · G · B64ᵀ|     B64: 64x160 complex (cropped 160-pt inverse DFT)
//   zh  = |C128· G · C128ᵀ|    C128: 128x160 complex (cropped 321-pt inverse DFT)
//   out = conv_w ⊗ low  +  pixel_unshuffle2(x - zh)
//
// Block = 128 threads = 4 wave32s computing a 32x32 output tile. Operand
// tiles are double-buffered in LDS in K-chunks of 32 and filled with the
// CDNA5 async global->LDS engine (ASYNCcnt / s_wait_asynccnt) when the
// toolchain exposes it, overlapping DMA with the WMMA stream; otherwise a
// synchronous ds_store staging path is used. Whole scratch (~120 MB) stays
// resident in the 192 MB L2.
// ---------------------------------------------------------------------------

typedef __attribute__((ext_vector_type(2))) float v2f;
typedef __attribute__((ext_vector_type(8))) float v8f;

#define NB   8      // batch
#define NC   32     // channels
#define NH   128    // H == W
#define NIMG (NB * NC)
#define NP   321    // padded size
#define NF   160    // cropped spectrum size
#define NL   64     // low-path output crop

// ---- workspace layout (floats). Complex stored planar: [re plane][im plane].
#define OFF_A    0                                   // 2*160*128
#define OFF_B64  (OFF_A   + 2 * NF * NH)             // 2*64*160
#define OFF_C128 (OFF_B64 + 2 * NL * NF)             // 2*128*160
#define OFF_G    (OFF_C128 + 2 * NH * NF)            // per img 2*160*160
#define OFF_P    (OFF_G + NIMG * 2 * NF * NF)        // per img 2*160*128 (reused by R)
#define OFF_Q    (OFF_P + NIMG * 2 * NF * NH)        // per img 2*64*160
#define OFF_L    (OFF_Q + NIMG * 2 * NL * NF)        // per img 64*64 (real)

#define SP 1056  // LDS plane stride: 32 rows x 33 floats (bank-conflict pad)

#if defined(__HIP_DEVICE_COMPILE__) &&                                   \
    __has_builtin(__builtin_amdgcn_global_load_async_to_lds_b32) &&      \
    __has_builtin(__builtin_amdgcn_s_wait_asynccnt)
#define FLC_ASYNC 1
#else
#define FLC_ASYNC 0
#endif

// One dword per lane, global -> LDS, via the async DMA path when available.
// (builtin is declared with int* params: global addrspace(1), LDS addrspace(3))
__device__ __forceinline__ void async_cp_b32(const float* g, float* l) {
#if FLC_ASYNC
  __builtin_amdgcn_global_load_async_to_lds_b32(
      (__attribute__((address_space(1))) int*)g,
      (__attribute__((address_space(3))) int*)l, 0, 0);
#else
  *l = *g;
  __builtin_prefetch(g + 32, 0, 3);  // stream next K-chunk toward L2/WGP$
#endif
}
template <int N>
__device__ __forceinline__ void async_wait() {
#if FLC_ASYNC
  __builtin_amdgcn_s_wait_asynccnt((unsigned short)N);
#endif
}

__device__ __forceinline__ v8f wmma4(v2f a, v2f b, v8f c) {
  // D = A(16x4,f32) * B(4x16,f32) + C(16x16,f32)
  return __builtin_amdgcn_wmma_f32_16x16x4_f32(false, a, false, b, (short)0, c,
                                               false, false);
}

// A-fragment: 16x4 tile at (r0, k0) of row-major [*, ld] (LDS).
// ISA layout: lane L holds row r0+(L&15); VGPR v holds K = k0 + (L>>4)*2 + v.
__device__ __forceinline__ v2f ldA(const float* p, int ld, int r0, int k0) {
  const int l = threadIdx.x & 31;
  const float* q = p + (r0 + (l & 15)) * ld + k0 + ((l >> 4) << 1);
  v2f r = {q[0], q[1]};
  return r;
}
// B-fragment: 4x16 tile at (k0, c0) of row-major [*, ld].
__device__ __forceinline__ v2f ldB(const float* p, int ld, int k0, int c0) {
  const int l = threadIdx.x & 31;
  const int kb = k0 + ((l >> 4) << 1);
  const int c = c0 + (l & 15);
  v2f r = {p[kb * ld + c], p[(kb + 1) * ld + c]};
  return r;
}
// D store: 16x16 tile at (r0, c0). VGPR v -> row r0 + v + (L>>4)*8, col L&15.
__device__ __forceinline__ void stD(float* __restrict__ p, int ld, int r0,
                                    int c0, v8f d) {
  const int l = threadIdx.x & 31;
  const int r = r0 + ((l >> 4) << 3);
  const int c = c0 + (l & 15);
#pragma unroll
  for (int v = 0; v < 8; ++v) p[(r + v) * ld + c] = d[v];
}

// ---------------------------------------------------------------------------
// Staging: one 32-wide K-chunk of the 32x32 tile's operands into 4 (or 3/2)
// padded LDS planes. 8 loop iterations x planes = async ops per thread.
// ---------------------------------------------------------------------------
template <bool BT>
__device__ __forceinline__ void stage_cc(float* dst, const float* Ar,
                                         const float* Ai, int lda,
                                         const float* Br, const float* Bi,
                                         int ldb, int m0, int n0, int kb) {
  for (int idx = threadIdx.x; idx < 1024; idx += 128) {
    const int r = idx >> 5, c = idx & 31;
    float* d = dst + r * 33 + c;
    async_cp_b32(Ar + (m0 + r) * lda + kb + c, d);
    async_cp_b32(Ai + (m0 + r) * lda + kb + c, d + SP);
    if (BT) {
      async_cp_b32(Br + (n0 + c) * ldb + kb + r, d + 2 * SP);
      async_cp_b32(Bi + (n0 + c) * ldb + kb + r, d + 3 * SP);
    } else {
      async_cp_b32(Br + (kb + r) * ldb + n0 + c, d + 2 * SP);
      async_cp_b32(Bi + (kb + r) * ldb + n0 + c, d + 3 * SP);
    }
  }
}
__device__ __forceinline__ void stage_cr(float* dst, const float* Ar,
                                         const float* Ai, int lda,
                                         const float* Br, int ldb, int m0,
                                         int n0, int kb) {
  for (int idx = threadIdx.x; idx < 1024; idx += 128) {
    const int r = idx >> 5, c = idx & 31;
    float* d = dst + r * 33 + c;
    async_cp_b32(Ar + (m0 + r) * lda + kb + c, d);
    async_cp_b32(Ai + (m0 + r) * lda + kb + c, d + SP);
    async_cp_b32(Br + (kb + r) * ldb + n0 + c, d + 2 * SP);
  }
}

// ---------------------------------------------------------------------------
// Complex 32x32-tile GEMM, double-buffered async LDS staging.
// BT: B is Mᵀ with M row-major [N, ldb].   Result: this wave's 16x16 tile.
// ---------------------------------------------------------------------------
template <bool BT>
__device__ __forceinline__ void cgemm32(const float* __restrict__ Ar,
                                        const float* __restrict__ Ai, int lda,
                                        const float* __restrict__ Br,
                                        const float* __restrict__ Bi, int ldb,
                                        int K, int m0, int n0, v8f& oR,
                                        v8f& oI) {
  __shared__ float sm[2][4 * SP];
  const int w = threadIdx.x >> 5, wm = w >> 1, wn = w & 1;
  v8f aR = {}, aI = {};
  const int nch = K >> 5;
  stage_cc<BT>(sm[0], Ar, Ai, lda, Br, Bi, ldb, m0, n0, 0);
  for (int i = 0; i < nch; ++i) {
    const int p = i & 1;
    if (i + 1 < nch) {  // prefetch next chunk into the other buffer
      stage_cc<BT>(sm[1 - p], Ar, Ai, lda, Br, Bi, ldb, m0, n0, (i + 1) << 5);
      async_wait<32>();  // in-order completion: chunk i (32 older ops) done
    } else {
      async_wait<0>();
    }
    __syncthreads();  // chunk i visible to all 4 waves
    const float* pA = sm[p];
#pragma unroll
    for (int kk = 0; kk < 32; kk += 4) {
      v2f ar = ldA(pA, 33, wm * 16, kk);
      v2f ai = ldA(pA + SP, 33, wm * 16, kk);
      v2f nai = {-ai[0], -ai[1]};
      v2f br = ldB(pA + 2 * SP, 33, kk, wn * 16);
      v2f bi = ldB(pA + 3 * SP, 33, kk, wn * 16);
      aR = wmma4(ar, br, aR);   // Re += Ar*Br
      aR = wmma4(nai, bi, aR);  // Re -= Ai*Bi
      aI = wmma4(ar, bi, aI);   // Im += Ar*Bi
      aI = wmma4(ai, br, aI);   // Im += Ai*Br
    }
    __syncthreads();  // all waves done with buffer p before it is refilled
  }
  oR = aR;
  oI = aI;
}

// Complex-A x real-B variant (stage 1: A(cplx) * x(real)).
__device__ __forceinline__ void crgemm32(const float* __restrict__ Ar,
                                         const float* __restrict__ Ai, int lda,
                                         const float* __restrict__ Br, int ldb,
                                         int K, int m0, int n0, v8f& oR,
                                         v8f& oI) {
  __shared__ float sm[2][3 * SP];
  const int w = threadIdx.x >> 5, wm = w >> 1, wn = w & 1;
  v8f aR = {}, aI = {};
  const int nch = K >> 5;
  stage_cr(sm[0], Ar, Ai, lda, Br, ldb, m0, n0, 0);
  for (int i = 0; i < nch; ++i) {
    const int p = i & 1;
    if (i + 1 < nch) {
      stage_cr(sm[1 - p], Ar, Ai, lda, Br, ldb, m0, n0, (i + 1) << 5);
      async_wait<24>();  // 24 newer ops in flight -> chunk i complete
    } else {
      async_wait<0>();
    }
    __syncthreads();
    const float* pA = sm[p];
#pragma unroll
    for (int kk = 0; kk < 32; kk += 4) {
      v2f ar = ldA(pA, 33, wm * 16, kk);
      v2f ai = ldA(pA + SP, 33, wm * 16, kk);
      v2f b = ldB(pA + 2 * SP, 33, kk, wn * 16);
      aR = wmma4(ar, b, aR);
      aI = wmma4(ai, b, aI);
    }
    __syncthreads();
  }
  oR = aR;
  oI = aI;
}

// Real x real variant (channel-mix conv GEMM), K = 32 (single chunk).
__device__ __forceinline__ void rgemm32(const float* __restrict__ Ap, int lda,
                                        const float* __restrict__ Bp, int ldb,
                                        int m0, int n0, v8f& oAcc) {
  __shared__ float sm[2 * SP];
  const int w = threadIdx.x >> 5, wm = w >> 1, wn = w & 1;
  for (int idx = threadIdx.x; idx < 1024; idx += 128) {
    const int r = idx >> 5, c = idx & 31;
    async_cp_b32(Ap + (m0 + r) * lda + c, sm + r * 33 + c);
    async_cp_b32(Bp + r * ldb + n0 + c, sm + SP + r * 33 + c);
  }
  async_wait<0>();
  __syncthreads();
  v8f acc = {};
#pragma unroll
  for (int kk = 0; kk < 32; kk += 4) {
    v2f a = ldA(sm, 33, wm * 16, kk);
    v2f b = ldB(sm + SP, 33, kk, wn * 16);
    acc = wmma4(a, b, acc);
  }
  oAcc = acc;
}

// ---------------------------------------------------------------------------
// K0: build the three DFT operator matrices with exact integer phase reduction.
// ---------------------------------------------------------------------------
__global__ void k_build_mats(float* __restrict__ ws) {
  const float PI2 = 6.283185307179586f;
  int t = blockIdx.x * blockDim.x + threadIdx.x;
  if (t < NF * NH) {  // A[r][m], freq f = r-80, reflect-pad fold, scale 1/321
    const int r = t / NH, m = t % NH;
    const int f = r - 80;
    int ns[3];
    int cnt = 0;
    ns[cnt++] = m + 97;                           // interior
    if (m >= 1 && m <= 97) ns[cnt++] = 97 - m;    // left reflection
    if (m >= 31 && m <= 126) ns[cnt++] = 351 - m; // right reflection
    float re = 0.f, im = 0.f;
    for (int i = 0; i < cnt; ++i) {
      int kk = (f * ns[i]) % NP;
      if (kk < 0) kk += NP;
      float s, c;
      sincosf(-PI2 * (float)kk / (float)NP, &s, &c);
      re += c;
      im += s;
    }
    const float sc = 1.0f / (float)NP;  // 1/321 per axis -> 1/321² total
    ws[OFF_A + t] = re * sc;
    ws[OFF_A + NF * NH + t] = im * sc;
    return;
  }
  t -= NF * NH;
  if (t < NL * NF) {  // B64[i][s] = exp(+2πi (48+i)(s-80)/160)
    const int i = t / NF, s = t % NF;
    int kk = ((48 + i) * (s - 80)) % NF;
    if (kk < 0) kk += NF;
    float sn, cs;
    sincosf(PI2 * (float)kk / (float)NF, &sn, &cs);
    ws[OFF_B64 + t] = cs;
    ws[OFF_B64 + NL * NF + t] = sn;
    return;
  }
  t -= NL * NF;
  if (t < NH * NF) {  // C128[i][s] = exp(+2πi (96+i)(s-80)/321)
    const int i = t / NF, s = t % NF;
    int kk = ((96 + i) * (s - 80)) % NP;
    if (kk < 0) kk += NP;
    float sn, cs;
    sincosf(PI2 * (float)kk / (float)NP, &sn, &cs);
    ws[OFF_C128 + t] = cs;
    ws[OFF_C128 + NH * NF + t] = sn;
  }
}

// ---------------------------------------------------------------------------
// K1: P = A(160x128, cplx) * x(128x128, real)   -> 160x128 cplx, per image
// ---------------------------------------------------------------------------
__global__ void k_stage1(const float* __restrict__ x,
                         const float* __restrict__ ws, float* __restrict__ Pb) {
  const int img = blockIdx.z;
  const int m0 = blockIdx.y << 5, n0 = blockIdx.x << 5;
  const float* Ar = ws + OFF_A;
  const float* Ai = Ar + NF * NH;
  const float* X = x + img * NH * NH;
  float* Pre = Pb + img * 2 * NF * NH;
  float* Pim = Pre + NF * NH;
  v8f aR, aI;
  crgemm32(Ar, Ai, NH, X, NH, NH, m0, n0, aR, aI);
  const int w = threadIdx.x >> 5;
  const int r0 = m0 + ((w >> 1) << 4), c0 = n0 + ((w & 1) << 4);
  stD(Pre, NH, r0, c0, aR);
  stD(Pim, NH, r0, c0, aI);
}

// ---------------------------------------------------------------------------
// K2: G = P(160x128) * Aᵀ(128x160)   complex x complex, per image
// ---------------------------------------------------------------------------
__global__ void k_stage2(const float* __restrict__ Pb,
                         const float* __restrict__ ws, float* __restrict__ Gb) {
  const int img = blockIdx.z;
  const int m0 = blockIdx.y << 5, n0 = blockIdx.x << 5;
  const float* Pre = Pb + img * 2 * NF * NH;
  const float* Pim = Pre + NF * NH;
  const float* Ar = ws + OFF_A;
  const float* Ai = Ar + NF * NH;
  float* Gre = Gb + img * 2 * NF * NF;
  float* Gim = Gre + NF * NF;
  v8f aR, aI;
  cgemm32<true>(Pre, Pim, NH, Ar, Ai, NH, NH, m0, n0, aR, aI);
  const int w = threadIdx.x >> 5;
  const int r0 = m0 + ((w >> 1) << 4), c0 = n0 + ((w & 1) << 4);
  stD(Gre, NF, r0, c0, aR);
  stD(Gim, NF, r0, c0, aI);
}

// ---------------------------------------------------------------------------
// K3: Q = B64(64x160) * G(160x160)   complex x complex, per image
// ---------------------------------------------------------------------------
__global__ void k_stage3(const float* __restrict__ Gb,
                         const float* __restrict__ ws, float* __restrict__ Qb) {
  const int img = blockIdx.z;
  const int m0 = blockIdx.y << 5, n0 = blockIdx.x << 5;
  const float* Br = ws + OFF_B64;
  const float* Bi = Br + NL * NF;
  const float* Gre = Gb + img * 2 * NF * NF;
  const float* Gim = Gre + NF * NF;
  float* Qre = Qb + img * 2 * NL * NF;
  float* Qim = Qre + NL * NF;
  v8f aR, aI;
  cgemm32<false>(Br, Bi, NF, Gre, Gim, NF, NF, m0, n0, aR, aI);
  const int w = threadIdx.x >> 5;
  const int r0 = m0 + ((w >> 1) << 4), c0 = n0 + ((w & 1) << 4);
  stD(Qre, NF, r0, c0, aR);
  stD(Qim, NF, r0, c0, aI);
}

// ---------------------------------------------------------------------------
// K4: Labs = | Q(64x160) * B64ᵀ(160x64) |   -> real 64x64, per image
// ---------------------------------------------------------------------------
__global__ void k_stage4(const float* __restrict__ Qb,
                         const float* __restrict__ ws, float* __restrict__ Lb) {
  const int img = blockIdx.z;
  const int m0 = blockIdx.y << 5, n0 = blockIdx.x << 5;
  const float* Qre = Qb + img * 2 * NL * NF;
  const float* Qim = Qre + NL * NF;
  const float* Br = ws + OFF_B64;
  const float* Bi = Br + NL * NF;
  float* L = Lb + img * NL * NL;
  v8f aR, aI;
  cgemm32<true>(Qre, Qim, NF, Br, Bi, NF, NF, m0, n0, aR, aI);
  const int w = threadIdx.x >> 5;
  const int l = threadIdx.x & 31;
  const int r = m0 + ((w >> 1) << 4) + ((l >> 4) << 3);
  const int c = n0 + ((w & 1) << 4) + (l & 15);
#pragma unroll
  for (int v = 0; v < 8; ++v)
    L[(r + v) * NL + c] = sqrtf(aR[v] * aR[v] + aI[v] * aI[v]);
}

// ---------------------------------------------------------------------------
// K7: out[b] = conv_w(128x32) * Labs[b](32x4096)   (real WMMA GEMM, writes out)
// ---------------------------------------------------------------------------
__global__ void k_conv(const float* __restrict__ Lb, const float* __restrict__ w,
                       float* __restrict__ out) {
  const int b = blockIdx.z;
  const int m0 = blockIdx.y << 5, n0 = blockIdx.x << 5;
  const float* Bm = Lb + b * NC * (NL * NL);  // (32 x 4096) row-major
  v8f acc;
  rgemm32(w, NC, Bm, NL * NL, m0, n0, acc);
  const int wv = threadIdx.x >> 5;
  const int r0 = m0 + ((wv >> 1) << 4), c0 = n0 + ((wv & 1) << 4);
  stD(out + b * (4 * NC) * (NL * NL), NL * NL, r0, c0, acc);
}

// ---------------------------------------------------------------------------
// K5: R = C128(128x160) * G(160x160)   complex x complex, per image
// ---------------------------------------------------------------------------
__global__ void k_stage5(const float* __restrict__ Gb,
                         const float* __restrict__ ws, float* __restrict__ Rb) {
  const int img = blockIdx.z;
  const int m0 = blockIdx.y << 5, n0 = blockIdx.x << 5;
  const float* Cr = ws + OFF_C128;
  const float* Ci = Cr + NH * NF;
  const float* Gre = Gb + img * 2 * NF * NF;
  const float* Gim = Gre + NF * NF;
  float* Rre = Rb + img * 2 * NH * NF;
  float* Rim = Rre + NH * NF;
  v8f aR, aI;
  cgemm32<false>(Cr, Ci, NF, Gre, Gim, NF, NF, m0, n0, aR, aI);
  const int w = threadIdx.x >> 5;
  const int r0 = m0 + ((w >> 1) << 4), c0 = n0 + ((w & 1) << 4);
  stD(Rre, NF, r0, c0, aR);
  stD(Rim, NF, r0, c0, aI);
}

// ---------------------------------------------------------------------------
// K6: Z = R(128x160) * C128ᵀ(160x128); zh=|Z|;
//     out[b, 4c+2(i&1)+(j&1), i/2, j/2] += x[b,c,i,j] - zh[i,j]
// ---------------------------------------------------------------------------
__global__ void k_stage6(const float* __restrict__ Rb,
                         const float* __restrict__ ws,
                         const float* __restrict__ x, float* __restrict__ out) {
  const int img = blockIdx.z;
  const int m0 = blockIdx.y << 5, n0 = blockIdx.x << 5;
  const float* Rre = Rb + img * 2 * NH * NF;
  const float* Rim = Rre + NH * NF;
  const float* Cr = ws + OFF_C128;
  const float* Ci = Cr + NH * NF;
  v8f aR, aI;
  cgemm32<true>(Rre, Rim, NF, Cr, Ci, NF, NF, m0, n0, aR, aI);
  const int w = threadIdx.x >> 5;
  const int l = threadIdx.x & 31;
  const int b = img >> 5, c = img & 31;
  const int i0 = m0 + ((w >> 1) << 4) + ((l >> 4) << 3);
  const int j = n0 + ((w & 1) << 4) + (l & 15);
#pragma unroll
  for (int v = 0; v < 8; ++v) {
    const int i = i0 + v;
    const float zh = sqrtf(aR[v] * aR[v] + aI[v] * aI[v]);
    const int oc = (c << 2) + ((i & 1) << 1) + (j & 1);
    out[(((b << 7) + oc) << 12) + ((i >> 1) << 6) + (j >> 1)] +=
        x[img * (NH * NH) + (i << 7) + j] - zh;
  }
}

// ---------------------------------------------------------------------------
extern "C" void kernel_launch(void* const* d_in, const int* in_sizes, int n_in,
                              void* d_out, int out_size, void* d_ws,
                              size_t ws_size, hipStream_t stream) {
  const float* x = (const float*)d_in[0];       // (8,32,128,128) f32
  const float* conv_w = (const float*)d_in[1];  // (128,32) f32
  float* out = (float*)d_out;                   // (8,128,64,64) f32
  float* ws = (float*)d_ws;

  float* P = ws + OFF_P;
  float* G = ws + OFF_G;
  float* Q = ws + OFF_Q;
  float* L = ws + OFF_L;
  float* R = ws + OFF_P;  // P is dead after stage 2; reuse its region for R

  const int nmat = NF * NH + NL * NF + NH * NF;
  k_build_mats<<<(nmat + 255) / 256, 256, 0, stream>>>(ws);

  // 4 wave32s per block, 32x32 tile per block; grid = (Ntiles, Mtiles, image)
  k_stage1<<<dim3(NH / 32, NF / 32, NIMG), 128, 0, stream>>>(x, ws, P);
  k_stage2<<<dim3(NF / 32, NF / 32, NIMG), 128, 0, stream>>>(P, ws, G);
  k_stage3<<<dim3(NF / 32, NL / 32, NIMG), 128, 0, stream>>>(G, ws, Q);
  k_stage4<<<dim3(NL / 32, NL / 32, NIMG), 128, 0, stream>>>(Q, ws, L);
  k_conv<<<dim3((NL * NL) / 32, (4 * NC) / 32, NB), 128, 0, stream>>>(L, conv_w,
                                                                      out);
  k_stage5<<<dim3(NF / 32, NH / 32, NIMG), 128, 0, stream>>>(G, ws, R);
  k_stage6<<<dim3(NH / 32, NH / 32, NIMG), 128, 0, stream>>>(R, ws, x, out);
}